// ContextualCentroidPerception_38250978738801
// MI455X (gfx1250) — compile-verified
//
#include <hip/hip_runtime.h>
#include <hip/hip_bf16.h>

// ---------------------------------------------------------------------------
// ContextualCentroidPerception for MI455X (gfx1250, wave32, WMMA + TDM)
// fp32 GEMMs via V_WMMA_F32_16X16X4_F32; feature tiles staged to LDS with the
// Tensor Data Mover; deterministic radix-select top-K.
// ---------------------------------------------------------------------------

typedef __attribute__((ext_vector_type(2))) float    v2f;
typedef __attribute__((ext_vector_type(8))) float    v8f;
typedef __attribute__((ext_vector_type(4))) unsigned u32x4;
typedef __attribute__((ext_vector_type(8))) int      i32x8;
typedef __attribute__((ext_vector_type(4))) int      i32x4;

typedef __attribute__((address_space(3))) float lds_float;

constexpr int B_   = 4;
constexpr int N_   = 32768;
constexpr int C_   = 256;
constexpr int K_   = 8192;
constexpr int MID_ = 128;
constexpr int NT_  = 32;          // points per block in conf head
constexpr float EPS_ = 1e-5f;

// ---------------------------------------------------------------------------
// Kernel 1: confidence head.
// Block = 32 points x all 256 output channels. The 256x32 fp32 feature tile
// (32 KB) is DMA'd to LDS once per block via TENSOR_LOAD_TO_LDS. 8 waves;
// wave w computes a 2x2 grid of 16x16 WMMA tiles: h rows {16w, 16(w+8)} x
// point halves {0,1}, K-reduction over C=256 in 64 steps of
// v_wmma_f32_16x16x4_f32. BN+ReLU -> LDS (aliasing the feature tile),
// 3x256 projection -> cls_preds, conf = sigmoid(max logit).
// ---------------------------------------------------------------------------
__global__ __launch_bounds__(256) void conf_head_kernel(
    const float* __restrict__ feats,   // (B,C,N)
    const float* __restrict__ w1,      // (256,256)
    const float* __restrict__ g1, const float* __restrict__ be1,
    const float* __restrict__ me1, const float* __restrict__ va1,
    const float* __restrict__ w2,      // (3,256)
    float* __restrict__ cls_out,       // (B,3,N)
    float* __restrict__ conf_out)      // (B,N)
{
    const int n0   = blockIdx.x * NT_;
    const int b    = blockIdx.y;
    const int tid  = threadIdx.x;
    const int lane = tid & 31;
    const int wave = tid >> 5;
    const int row16 = lane & 15;      // M row (A) / N col (B,C/D)
    const int khalf = lane >> 4;      // selects K pair {0,1} vs {2,3}

    // smem holds the feature tile [c][nt] during the GEMM, then (after a
    // barrier) is reused for the h tile [o][nt].  Both are 256*32 floats.
    __shared__ float smem[C_ * NT_];  // 32 KB
    __shared__ float clslds[3 * NT_];

    // ---- Tensor Data Mover: features[b, 0:256, n0:n0+32] -> LDS ----------
    if (wave == 0) {
        const float* gsrc = feats + (size_t)b * C_ * N_ + n0;
        unsigned long long ga = (unsigned long long)(size_t)gsrc;
        unsigned lds_off = (unsigned)(size_t)(lds_float*)smem;

        u32x4 g0;
        g0.x = 1u;                                    // count=1, no gather
        g0.y = lds_off;                               // D#.lds_addr
        g0.z = (unsigned)(ga & 0xFFFFFFFFull);        // global_addr[31:0]
        g0.w = (unsigned)((ga >> 32) & 0x01FFFFFFull) // global_addr[56:32]
             | (2u << 30);                            // type = 2 (image)

        i32x8 g1v;
        g1v[0] = 0x00020000;                          // data_size = 4 bytes
        g1v[1] = (int)((N_ & 0xFFFF) << 16);          // tensor_dim0[15:0]
        g1v[2] = (int)(((unsigned)N_ >> 16) | ((C_ & 0xFFFF) << 16)); // dim0 hi | dim1 lo
        g1v[3] = (int)(NT_ << 16);                    // dim1 hi=0 | tile_dim0=32
        g1v[4] = C_;                                  // tile_dim1=256, tile_dim2=0
        g1v[5] = N_;                                  // tensor_dim0_stride[31:0]
        g1v[6] = 0;                                   // stride hi | dim1_stride lo
        g1v[7] = 0;
        i32x4 z4 = {0, 0, 0, 0};
        i32x8 z8 = {0, 0, 0, 0, 0, 0, 0, 0};
        __builtin_amdgcn_tensor_load_to_lds(g0, g1v, z4, z4, z8, 0);
        __builtin_amdgcn_s_wait_tensorcnt(0);
    }
    __syncthreads();

    v8f acc00 = {}, acc01 = {};       // rows o0, point halves 0/1
    v8f acc10 = {}, acc11 = {};       // rows o1
    const int o0 = wave * 16;
    const int o1 = (wave + 8) * 16;

    for (int k0 = 0; k0 < C_; k0 += 4) {
        const int kk = k0 + 2 * khalf;
        // B operands from the LDS feature tile
        v2f b0, b1;
        b0.x = smem[kk * NT_ + row16];
        b0.y = smem[(kk + 1) * NT_ + row16];
        b1.x = smem[kk * NT_ + 16 + row16];
        b1.y = smem[(kk + 1) * NT_ + 16 + row16];
        // A operands: 16x4 weight tiles (L2-resident)
        v2f a0 = *(const v2f*)(w1 + (size_t)(o0 + row16) * C_ + kk);
        v2f a1 = *(const v2f*)(w1 + (size_t)(o1 + row16) * C_ + kk);
        acc00 = __builtin_amdgcn_wmma_f32_16x16x4_f32(
            false, a0, false, b0, (short)0, acc00, false, false);
        acc01 = __builtin_amdgcn_wmma_f32_16x16x4_f32(
            false, a0, false, b1, (short)0, acc01, false, false);
        acc10 = __builtin_amdgcn_wmma_f32_16x16x4_f32(
            false, a1, false, b0, (short)0, acc10, false, false);
        acc11 = __builtin_amdgcn_wmma_f32_16x16x4_f32(
            false, a1, false, b1, (short)0, acc11, false, false);
    }
    __syncthreads();   // everyone done reading the feature tile

    // BN + ReLU, stage h into smem (aliased).  VGPR v holds M = v + 8*khalf.
    #pragma unroll
    for (int v = 0; v < 8; ++v) {
        int oA = o0 + v + 8 * khalf;
        float invA = g1[oA] * rsqrtf(va1[oA] + EPS_);
        float addA = be1[oA] - me1[oA] * invA;
        smem[oA * NT_ + row16]      = fmaxf(acc00[v] * invA + addA, 0.0f);
        smem[oA * NT_ + 16 + row16] = fmaxf(acc01[v] * invA + addA, 0.0f);
        int oB = o1 + v + 8 * khalf;
        float invB = g1[oB] * rsqrtf(va1[oB] + EPS_);
        float addB = be1[oB] - me1[oB] * invB;
        smem[oB * NT_ + row16]      = fmaxf(acc10[v] * invB + addB, 0.0f);
        smem[oB * NT_ + 16 + row16] = fmaxf(acc11[v] * invB + addB, 0.0f);
    }
    __syncthreads();

    // 3x256 projection -> class logits
    if (tid < 96) {
        int c = tid >> 5, nl = tid & 31;
        float s = 0.0f;
        #pragma unroll 8
        for (int k = 0; k < C_; ++k) s += w2[c * C_ + k] * smem[k * NT_ + nl];
        cls_out[((size_t)b * 3 + c) * N_ + n0 + nl] = s;
        clslds[c * NT_ + nl] = s;
    }
    __syncthreads();

    if (tid < NT_) {
        float mx = fmaxf(clslds[tid],
                         fmaxf(clslds[NT_ + tid], clslds[2 * NT_ + tid]));
        conf_out[(size_t)b * N_ + n0 + tid] = 1.0f / (1.0f + __expf(-mx));
    }
}

// ---------------------------------------------------------------------------
// Kernel 2: per-batch 4-pass MSB radix select of the K-th largest confidence.
// conf = sigmoid(.) > 0, so the raw fp32 bit pattern is uint-monotone.
// Outputs: T[b] = exact key of K-th largest, G[b] = #strictly greater.
// ---------------------------------------------------------------------------
__global__ __launch_bounds__(256) void topk_select_kernel(
    const float* __restrict__ conf, unsigned* __restrict__ Tout,
    int* __restrict__ Gout)
{
    const int b = blockIdx.x;
    const int tid = threadIdx.x;
    __shared__ unsigned hist[256];
    __shared__ unsigned s_prefix;
    __shared__ int s_rem;
    if (tid == 0) { s_prefix = 0u; s_rem = K_; }
    __syncthreads();

    for (int pass = 0; pass < 4; ++pass) {
        const int shift = 24 - 8 * pass;
        const unsigned pmask = (pass == 0) ? 0u : (0xFFFFFFFFu << (32 - 8 * pass));
        hist[tid] = 0u;
        __syncthreads();
        for (int nn = tid; nn < N_; nn += 256) {
            unsigned u = __float_as_uint(conf[(size_t)b * N_ + nn]);
            if ((u & pmask) == s_prefix) atomicAdd(&hist[(u >> shift) & 255], 1u);
        }
        __syncthreads();
        if (tid == 0) {
            int rem = s_rem;
            unsigned pref = s_prefix;
            for (int bin = 255; bin >= 0; --bin) {
                int c = (int)hist[bin];
                if (c >= rem) { s_prefix = pref | ((unsigned)bin << shift); break; }
                rem -= c;
            }
            s_rem = rem;
        }
        __syncthreads();
    }
    if (tid == 0) { Tout[b] = s_prefix; Gout[b] = K_ - s_rem; }
}

// ---------------------------------------------------------------------------
// Kernel 3: deterministic compaction via block-wide scans (no atomic slots).
// Strictly-greater entries take [0,G) in index order; exactly K-G ties take
// [G,K) in index order.
// ---------------------------------------------------------------------------
__global__ __launch_bounds__(256) void topk_compact_kernel(
    const float* __restrict__ conf, const unsigned* __restrict__ Tin,
    const int* __restrict__ Gin, int* __restrict__ idx)
{
    const int b = blockIdx.x;
    const int tid = threadIdx.x;
    __shared__ int sg[256], se[256];
    __shared__ int baseG, baseE;
    if (tid == 0) { baseG = 0; baseE = 0; }
    const unsigned T = Tin[b];
    const int G = Gin[b];
    __syncthreads();

    for (int chunk = 0; chunk < N_; chunk += 256) {
        int nn = chunk + tid;
        unsigned u = __float_as_uint(conf[(size_t)b * N_ + nn]);
        int fg = (u > T) ? 1 : 0;
        int fe = (u == T) ? 1 : 0;
        sg[tid] = fg; se[tid] = fe;
        __syncthreads();
        for (int off = 1; off < 256; off <<= 1) {
            int vg = (tid >= off) ? sg[tid - off] : 0;
            int ve = (tid >= off) ? se[tid - off] : 0;
            __syncthreads();
            sg[tid] += vg; se[tid] += ve;
            __syncthreads();
        }
        if (fg) { int p = baseG + sg[tid] - 1; if (p < K_) idx[(size_t)b * K_ + p] = nn; }
        if (fe) { int p = G + baseE + se[tid] - 1; if (p < K_) idx[(size_t)b * K_ + p] = nn; }
        __syncthreads();
        if (tid == 0) { baseG += sg[255]; baseE += se[255]; }
        __syncthreads();
    }
}

// ---------------------------------------------------------------------------
// Kernel 4: centroid regression head on gathered columns.
// Block = 16 gathered points x 128 MID rows (8 waves x one 16-row tile),
// WMMA f32 16x16x4 with gathered B columns, BN+ReLU -> LDS,
// 3x128 projection, clip, add origins.
// ---------------------------------------------------------------------------
__global__ __launch_bounds__(256) void reg_head_kernel(
    const float* __restrict__ feats,   // (B,C,N)
    const float* __restrict__ points,  // (B,N,3)
    const int* __restrict__ idx,       // (B,K)
    const float* __restrict__ wr1,     // (128,256)
    const float* __restrict__ g2, const float* __restrict__ be2,
    const float* __restrict__ me2, const float* __restrict__ va2,
    const float* __restrict__ wr2,     // (3,128)
    float* __restrict__ preds, float* __restrict__ origins,
    float* __restrict__ offsets)       // each (B,K,3)
{
    const int j0   = blockIdx.x * 16;
    const int b    = blockIdx.y;
    const int tid  = threadIdx.x;
    const int lane = tid & 31;
    const int wave = tid >> 5;
    const int row16 = lane & 15;
    const int khalf = lane >> 4;

    __shared__ float glds[MID_ * 16];  // 8 KB

    const float* fb = feats + (size_t)b * C_ * N_;
    const int col = idx[(size_t)b * K_ + j0 + row16];

    v8f acc = {};
    const int m0 = wave * 16;
    for (int k0 = 0; k0 < C_; k0 += 4) {
        const int kk = k0 + 2 * khalf;
        v2f bv;                                  // gathered feature columns
        bv.x = fb[(size_t)kk * N_ + col];
        bv.y = fb[(size_t)(kk + 1) * N_ + col];
        __builtin_prefetch(fb + (size_t)(kk + 4) * N_ + col, 0, 1);
        v2f a = *(const v2f*)(wr1 + (size_t)(m0 + row16) * C_ + kk);
        acc = __builtin_amdgcn_wmma_f32_16x16x4_f32(
            false, a, false, bv, (short)0, acc, false, false);
    }

    #pragma unroll
    for (int v = 0; v < 8; ++v) {
        int m = m0 + v + 8 * khalf;
        float inv = g2[m] * rsqrtf(va2[m] + EPS_);
        float val = acc[v] * inv + (be2[m] - me2[m] * inv);
        glds[m * 16 + row16] = fmaxf(val, 0.0f);
    }
    __syncthreads();

    if (tid < 48) {
        int o = tid >> 4, nl = tid & 15;
        int j = j0 + nl;
        int ii = idx[(size_t)b * K_ + j];
        float s = 0.0f;
        #pragma unroll 8
        for (int k = 0; k < MID_; ++k) s += wr2[o * MID_ + k] * glds[k * 16 + nl];
        float org = points[((size_t)b * N_ + ii) * 3 + o];
        float mo = (o == 2) ? 2.0f : 3.0f;
        float lim = fminf(fmaxf(s, -mo), mo);
        size_t base = ((size_t)b * K_ + j) * 3 + o;
        offsets[base] = s;          // unclipped, per reference
        origins[base] = org;
        preds[base]   = org + lim;
    }
}

// ---------------------------------------------------------------------------
extern "C" void kernel_launch(void* const* d_in, const int* in_sizes, int n_in,
                              void* d_out, int out_size, void* d_ws, size_t ws_size,
                              hipStream_t stream) {
    const float* points    = (const float*)d_in[0];
    const float* features  = (const float*)d_in[1];
    const float* w_conf1   = (const float*)d_in[2];
    const float* bn1_gamma = (const float*)d_in[3];
    const float* bn1_beta  = (const float*)d_in[4];
    const float* bn1_mean  = (const float*)d_in[5];
    const float* bn1_var   = (const float*)d_in[6];
    const float* w_conf2   = (const float*)d_in[7];
    const float* w_reg1    = (const float*)d_in[8];
    const float* bn2_gamma = (const float*)d_in[9];
    const float* bn2_beta  = (const float*)d_in[10];
    const float* bn2_mean  = (const float*)d_in[11];
    const float* bn2_var   = (const float*)d_in[12];
    const float* w_reg2    = (const float*)d_in[13];

    float* out      = (float*)d_out;
    float* preds    = out;                            // (B,K,3)
    float* origins  = out + (size_t)B_ * K_ * 3;      // (B,K,3)
    float* offsets  = out + (size_t)2 * B_ * K_ * 3;  // (B,K,3)
    float* cls_out  = out + (size_t)3 * B_ * K_ * 3;  // (B,3,N)

    float*    conf = (float*)d_ws;                    // B*N floats
    unsigned* Tws  = (unsigned*)(conf + (size_t)B_ * N_);
    int*      Gws  = (int*)(Tws + B_);
    int*      idxw = (int*)(Gws + B_);                // B*K ints

    conf_head_kernel<<<dim3(N_ / NT_, B_), 256, 0, stream>>>(
        features, w_conf1, bn1_gamma, bn1_beta, bn1_mean, bn1_var,
        w_conf2, cls_out, conf);

    topk_select_kernel<<<B_, 256, 0, stream>>>(conf, Tws, Gws);

    topk_compact_kernel<<<B_, 256, 0, stream>>>(conf, Tws, Gws, idxw);

    reg_head_kernel<<<dim3(K_ / 16, B_), 256, 0, stream>>>(
        features, points, idxw, w_reg1, bn2_gamma, bn2_beta, bn2_mean,
        bn2_var, w_reg2, preds, origins, offsets);
}